// Raymarcher_6536940224697
// MI455X (gfx1250) — compile-verified
//
#include <hip/hip_runtime.h>
#include <hip/hip_bf16.h>
#include <stdint.h>

#define VOLR   256.0f
#define NSTEPS 64
#define HDIM   128
#define WDIM   128
#define KPRIM  64
#define SDIM   16
#define NPIX   (HDIM * WDIM)
#define TABF   (KPRIM * 16)          // 1024 floats = 4 KB transform table
#define VOXK   (SDIM * SDIM * SDIM)  // 4096 voxels per primitive

// workspace layout (bytes)
#define TAB_OFF   0u
#define SPH_OFF   4096u                       // 64 prims x 8 floats = 2 KB
#define MASK_OFF  8192u                       // 16384 rays x 2 u32 = 128 KB
#define TPL_OFF   (MASK_OFF + 131072u)        // 4 MB transposed template
#define TPL_BYTES ((size_t)KPRIM * VOXK * 16u)

typedef __attribute__((ext_vector_type(2))) float        v2f;
typedef __attribute__((ext_vector_type(8))) float        v8f;
typedef __attribute__((ext_vector_type(4))) unsigned int u32x4;
typedef __attribute__((ext_vector_type(4))) int          i32x4;
typedef __attribute__((ext_vector_type(8))) int          i32x8;

// ---------------- small helpers ----------------
__device__ __forceinline__ float4 f4lerp(float4 a, float4 b, float t) {
    return make_float4(fmaf(t, b.x - a.x, a.x),
                       fmaf(t, b.y - a.y, a.y),
                       fmaf(t, b.z - a.z, a.z),
                       fmaf(t, b.w - a.w, a.w));
}

__device__ __forceinline__ uint32_t ballot32(bool c) {
#if defined(__gfx1250__) && __has_builtin(__builtin_amdgcn_ballot_w32)
    return __builtin_amdgcn_ballot_w32(c);
#else
    return (uint32_t)__ballot((int)c);
#endif
}

// ---------------- kernel 1: per-primitive affine table + bounding spheres ----------------
// tab per prim (16 floats): p[3], scale[3], Rcol0[3], Rcol1[3], Rcol2[3], pad
// sph per prim (8 floats):  w = p - vp [3], |w|^2, r2margin, pad[3]
__global__ void build_tab_kernel(const float* __restrict__ viewpos,
                                 const float* __restrict__ primpos,
                                 const float* __restrict__ primrot,
                                 const float* __restrict__ primscale,
                                 float* __restrict__ tab,
                                 float* __restrict__ sph) {
    int k = blockIdx.x * blockDim.x + threadIdx.x;
    if (k >= KPRIM) return;
    float* T = tab + k * 16;
    float p0 = primpos[k * 3 + 0], p1 = primpos[k * 3 + 1], p2 = primpos[k * 3 + 2];
    float s0 = primscale[k * 3 + 0], s1 = primscale[k * 3 + 1], s2 = primscale[k * 3 + 2];
    T[0] = p0; T[1] = p1; T[2] = p2;
    T[3] = s0; T[4] = s1; T[5] = s2;
#pragma unroll
    for (int j = 0; j < 3; ++j)
#pragma unroll
        for (int i = 0; i < 3; ++i)
            T[6 + 3 * j + i] = primrot[k * 9 + i * 3 + j];   // column j of R
    T[15] = 0.0f;

    // bounding sphere: local unit cube maps into |x-p| <= sqrt(sum 1/s_j^2)
    float wx = p0 - viewpos[0] / VOLR;
    float wy = p1 - viewpos[1] / VOLR;
    float wz = p2 - viewpos[2] / VOLR;
    float r2 = 1.0f / (s0 * s0) + 1.0f / (s1 * s1) + 1.0f / (s2 * s2);
    float* S = sph + k * 8;
    S[0] = wx; S[1] = wy; S[2] = wz;
    S[3] = wx * wx + wy * wy + wz * wz;
    S[4] = r2 * 1.001f + 1e-4f;                // conservative fp margin
    S[5] = 0.0f; S[6] = 0.0f; S[7] = 0.0f;
}

// ---------------- kernel 2: transpose template [K,4,S,S,S] -> float4 [K,S,S,S] ----------------
__global__ void transpose_tpl_kernel(const float* __restrict__ tpl,
                                     float4* __restrict__ tplT) {
    int v = blockIdx.x * blockDim.x + threadIdx.x;
    if (v >= KPRIM * VOXK) return;
    int k = v >> 12;
    int r = v & (VOXK - 1);
    const float* b = tpl + (size_t)k * 4 * VOXK + r;
    tplT[v] = make_float4(b[0], b[VOXK], b[2 * VOXK], b[3 * VOXK]);
}

// ---------------- kernel 3: WMMA ray-primitive culling masks ----------------
// One wave per 16 rays. C[r,k] = d_r . w_k via V_WMMA_F32_16X16X4_F32 tiles;
// mask bit k set iff line-sphere distance^2 < r^2 (conservative superset of hits).
__global__ __launch_bounds__(32) void mask_kernel(const float* __restrict__ raydir,
                                                  const float* __restrict__ sph,
                                                  uint32_t* __restrict__ maskbuf) {
    const int lane = (int)threadIdx.x;      // 0..31, one full wave32
    const int base = (int)blockIdx.x * 16;  // 16 rays per wave
    const int m = lane & 15;

    // A (16x4 f32): lanes 0-15 hold K0,K1 = (dx,dy); lanes 16-31 hold K2,K3 = (dz,0)
    const float* d = raydir + (size_t)(base + m) * 3;
    v2f a;
    a[0] = (lane < 16) ? d[0] : d[2];
    a[1] = (lane < 16) ? d[1] : 0.0f;

    uint32_t w0 = 0, w1 = 0;
#pragma unroll
    for (int g = 0; g < 4; ++g) {
        const float* S = sph + (size_t)(g * 16 + m) * 8;
        float ww2 = S[3];
        float r2m = S[4];
        v8f c = {0, 0, 0, 0, 0, 0, 0, 0};
#if defined(__gfx1250__) && __has_builtin(__builtin_amdgcn_wmma_f32_16x16x4_f32)
        // B (4x16 f32): VGPR0 = rows K0 (lanes 0-15) / K2 (lanes 16-31); VGPR1 = K1 / K3
        v2f b;
        b[0] = (lane < 16) ? S[0] : S[2];
        b[1] = (lane < 16) ? S[1] : 0.0f;
        c = __builtin_amdgcn_wmma_f32_16x16x4_f32(false, a, false, b,
                                                  (short)0, c, false, false);
#else
        // scalar fallback: lane owns prim column N = m, rows M = r + 8*(lane>=16)
#pragma unroll
        for (int r = 0; r < 8; ++r) {
            int ray = base + r + ((lane >= 16) ? 8 : 0);
            const float* dd = raydir + (size_t)ray * 3;
            c[r] = fmaf(dd[2], S[2], fmaf(dd[1], S[1], dd[0] * S[0]));
        }
#endif
        // D layout: VGPR r, lane L -> (ray = base + r + 8*(L>=16), prim = g*16 + L%16)
        uint32_t sel = 0;
#pragma unroll
        for (int r = 0; r < 8; ++r) {
            float cv = c[r];
            bool hit = (cv * cv) > (ww2 - r2m);   // dist^2 = |w|^2 - c^2 < r^2
            uint32_t bal = ballot32(hit);         // low 16: ray base+r, high 16: ray base+r+8
            if ((lane & 7) == r) sel = bal;
        }
        uint32_t bits = (sel >> (((lane >> 3) & 1) * 16)) & 0xFFFFu;
        if (g < 2) w0 |= bits << (16 * g);
        else       w1 |= bits << (16 * (g - 2));
    }
    if (lane < 16) {
        maskbuf[(size_t)(base + lane) * 2 + 0] = w0;
        maskbuf[(size_t)(base + lane) * 2 + 1] = w1;
    }
}

// ---------------- kernel 4: raymarch ----------------
__global__ __launch_bounds__(64) void raymarch_kernel(
    const float* __restrict__ viewpos,
    const float* __restrict__ raydir,
    const float* __restrict__ dtp,
    const float* __restrict__ tabG,        // TDM source (null if !useTdm)
    const float4* __restrict__ tplT,       // transposed template (valid if useT)
    const float* __restrict__ tplP,        // original planar template
    const float* __restrict__ primpos,     // fallback table build
    const float* __restrict__ primrot,
    const float* __restrict__ primscale,
    const uint32_t* __restrict__ maskbuf,  // per-ray prim masks (valid if useMask)
    float* __restrict__ out,
    int useT, int useTdm, int useMask)
{
    __shared__ float tab[TABF];

    bool filled = false;
#if defined(__gfx1250__) && __has_builtin(__builtin_amdgcn_tensor_load_to_lds)
    if (useTdm) {
        if (threadIdx.x < 32) {  // wave 0 issues one TDM op: 4 KB global -> LDS
            uint64_t ga = (uint64_t)(uintptr_t)tabG;
            uint32_t lo = (uint32_t)(uintptr_t)(&tab[0]);
            u32x4 g0;
            g0[0] = 1u;                                                // count=1
            g0[1] = lo;                                                // lds_addr
            g0[2] = (uint32_t)ga;                                      // global_addr lo
            g0[3] = (uint32_t)((ga >> 32) & 0x01FFFFFFu) | (2u << 30); // addr hi | type=2
            i32x8 g1;
            g1[0] = (int)(2u << 16);      // data_size = 4B
            g1[1] = (int)(1024u << 16);   // tensor_dim0 = 1024
            g1[2] = (int)(1u << 16);      // tensor_dim1 = 1
            g1[3] = (int)(1024u << 16);   // tile_dim0 = 1024
            g1[4] = 1;                    // tile_dim1 = 1
            g1[5] = 1024;                 // tensor_dim0_stride
            g1[6] = 0;
            g1[7] = 0;
            i32x4 z4 = {0, 0, 0, 0};
#if __clang_major__ >= 23
            i32x8 z8 = {0, 0, 0, 0, 0, 0, 0, 0};
            __builtin_amdgcn_tensor_load_to_lds(g0, g1, z4, z4, z8, 0);
#else
            __builtin_amdgcn_tensor_load_to_lds(g0, g1, z4, z4, 0);
#endif
            __builtin_amdgcn_s_wait_tensorcnt(0);
        }
        filled = true;
    }
#endif
    if (!filled) {
        for (int k = (int)threadIdx.x; k < KPRIM; k += (int)blockDim.x) {
            float* T = &tab[k * 16];
#pragma unroll
            for (int i = 0; i < 3; ++i) T[i]     = primpos[k * 3 + i];
#pragma unroll
            for (int i = 0; i < 3; ++i) T[3 + i] = primscale[k * 3 + i];
#pragma unroll
            for (int j = 0; j < 3; ++j)
#pragma unroll
                for (int i = 0; i < 3; ++i)
                    T[6 + 3 * j + i] = primrot[k * 9 + i * 3 + j];
            T[15] = 0.0f;
        }
    }
    __syncthreads();

    int pix = (int)(blockIdx.x * blockDim.x + threadIdx.x);

    __builtin_prefetch(tplP + (size_t)(pix & 1023) * 1024, 0, 0);

    float vx = viewpos[0] / VOLR, vy = viewpos[1] / VOLR, vz = viewpos[2] / VOLR;
    float dtn = dtp[0] / VOLR;
    float rx = raydir[3 * pix + 0], ry = raydir[3 * pix + 1], rz = raydir[3 * pix + 2];

    // ray / [-1,1]^3 slab intersection
    float t1x = (-1.0f - vx) / rx, t2x = (1.0f - vx) / rx;
    float t1y = (-1.0f - vy) / ry, t2y = (1.0f - vy) / ry;
    float t1z = (-1.0f - vz) / rz, t2z = (1.0f - vz) / rz;
    float tmn = fmaxf(fminf(t1x, t2x), fmaxf(fminf(t1y, t2y), fminf(t1z, t2z)));
    float tmx = fminf(fmaxf(t1x, t2x), fminf(fmaxf(t1y, t2y), fmaxf(t1z, t2z)));
    bool inter = tmn < tmx;
    float tmin_ = inter ? tmn : 0.0f;
    float tmax_ = inter ? tmx : 0.0f;
    float t0 = fmaxf(tmin_, 0.0f);

    uint32_t pm0 = 0xFFFFFFFFu, pm1 = 0xFFFFFFFFu;
    if (useMask) {
        pm0 = maskbuf[(size_t)pix * 2 + 0];
        pm1 = maskbuf[(size_t)pix * 2 + 1];
    }

    float ax = 0.0f, ay = 0.0f, az = 0.0f, aw = 0.0f;

    for (int i = 0; i < NSTEPS; ++i) {
        float t = fmaf((float)i, dtn, t0);
        if (!(t < tmax_)) break;
        float px = fmaf(rx, t, vx);
        float py = fmaf(ry, t, vy);
        float pz = fmaf(rz, t, vz);

        float sx = 0.0f, sy = 0.0f, sz = 0.0f, sw = 0.0f;

        auto test_prim = [&](int k) {
            const float* T = &tab[k * 16];       // LDS broadcast/indexed reads
            float dx = px - T[0], dy = py - T[1], dz = pz - T[2];
            float y0 = fmaf(dz, T[8],  fmaf(dy, T[7],  dx * T[6]))  * T[3];
            float y1 = fmaf(dz, T[11], fmaf(dy, T[10], dx * T[9]))  * T[4];
            float y2 = fmaf(dz, T[14], fmaf(dy, T[13], dx * T[12])) * T[5];
            if (fabsf(y0) < 1.0f && fabsf(y1) < 1.0f && fabsf(y2) < 1.0f) {
                const float HS = 0.5f * (float)(SDIM - 1);
                float gx = (y0 + 1.0f) * HS;
                float gy = (y1 + 1.0f) * HS;
                float gz = (y2 + 1.0f) * HS;
                float fx0 = fminf(fmaxf(floorf(gx), 0.0f), (float)(SDIM - 2));
                float fy0 = fminf(fmaxf(floorf(gy), 0.0f), (float)(SDIM - 2));
                float fz0 = fminf(fmaxf(floorf(gz), 0.0f), (float)(SDIM - 2));
                int ix = (int)fx0, iy = (int)fy0, iz = (int)fz0;
                float fx = gx - fx0, fy = gy - fy0, fz = gz - fz0;

                float4 c000, c001, c010, c011, c100, c101, c110, c111;
                if (useT) {
                    const float4* b = tplT + (((k * SDIM + iz) * SDIM + iy) * SDIM + ix);
                    c000 = b[0];                     c001 = b[1];
                    c010 = b[SDIM];                  c011 = b[SDIM + 1];
                    c100 = b[SDIM * SDIM];           c101 = b[SDIM * SDIM + 1];
                    c110 = b[SDIM * SDIM + SDIM];    c111 = b[SDIM * SDIM + SDIM + 1];
                } else {
                    const float* b0 = tplP + ((size_t)k * 4 * VOXK + (iz * SDIM + iy) * SDIM + ix);
                    auto ld = [&](int off) {
                        return make_float4(b0[off], b0[off + VOXK],
                                           b0[off + 2 * VOXK], b0[off + 3 * VOXK]);
                    };
                    c000 = ld(0);                    c001 = ld(1);
                    c010 = ld(SDIM);                 c011 = ld(SDIM + 1);
                    c100 = ld(SDIM * SDIM);          c101 = ld(SDIM * SDIM + 1);
                    c110 = ld(SDIM * SDIM + SDIM);   c111 = ld(SDIM * SDIM + SDIM + 1);
                }
                float4 c00 = f4lerp(c000, c001, fx);
                float4 c01 = f4lerp(c010, c011, fx);
                float4 c10 = f4lerp(c100, c101, fx);
                float4 c11 = f4lerp(c110, c111, fx);
                float4 c0  = f4lerp(c00, c01, fy);
                float4 c1  = f4lerp(c10, c11, fy);
                float4 c   = f4lerp(c0, c1, fz);
                sx += c.x; sy += c.y; sz += c.z; sw += c.w;
            }
        };

        for (uint32_t mm = pm0; mm; mm &= mm - 1) test_prim((int)__builtin_ctz(mm));
        for (uint32_t mm = pm1; mm; mm &= mm - 1) test_prim(32 + (int)__builtin_ctz(mm));

        float alpha   = sw * dtn;
        float contrib = fminf(aw + alpha, 1.0f) - aw;
        ax = fmaf(sx, contrib, ax);
        ay = fmaf(sy, contrib, ay);
        az = fmaf(sz, contrib, az);
        aw += contrib;
        if (aw >= 1.0f) break;
    }

    // outputs: rgba [1,4,H,W] ++ rayposbeg [1,H,W,3] ++ rayposend [1,H,W,3]
    out[0 * NPIX + pix] = ax;
    out[1 * NPIX + pix] = ay;
    out[2 * NPIX + pix] = az;
    out[3 * NPIX + pix] = aw;
    float* beg = out + 4 * NPIX;
    float* end = out + 7 * NPIX;
    beg[3 * pix + 0] = fmaf(rx, tmin_, vx);
    beg[3 * pix + 1] = fmaf(ry, tmin_, vy);
    beg[3 * pix + 2] = fmaf(rz, tmin_, vz);
    end[3 * pix + 0] = fmaf(rx, tmax_, vx);
    end[3 * pix + 1] = fmaf(ry, tmax_, vy);
    end[3 * pix + 2] = fmaf(rz, tmax_, vz);
}

// ---------------- host launcher ----------------
extern "C" void kernel_launch(void* const* d_in, const int* in_sizes, int n_in,
                              void* d_out, int out_size, void* d_ws, size_t ws_size,
                              hipStream_t stream) {
    const float* viewpos   = (const float*)d_in[0];
    // d_in[1] = viewrot (unused by reference)
    const float* raydir    = (const float*)d_in[2];
    const float* tpl       = (const float*)d_in[3];
    const float* primpos   = (const float*)d_in[4];
    const float* primrot   = (const float*)d_in[5];
    const float* primscale = (const float*)d_in[6];
    const float* dtp       = (const float*)d_in[7];

    float*    tab  = (float*)((char*)d_ws + TAB_OFF);
    float*    sph  = (float*)((char*)d_ws + SPH_OFF);
    uint32_t* mask = (uint32_t*)((char*)d_ws + MASK_OFF);
    float4*   tplT = (float4*)((char*)d_ws + TPL_OFF);

    int useTdm  = (ws_size >= SPH_OFF + 2048) ? 1 : 0;               // tab + sph present
    int useMask = (ws_size >= MASK_OFF + (size_t)NPIX * 8) ? 1 : 0;  // + 128 KB masks
    int useT    = (ws_size >= TPL_OFF + TPL_BYTES) ? 1 : 0;          // + 4 MB template

    if (useTdm)
        build_tab_kernel<<<1, 64, 0, stream>>>(viewpos, primpos, primrot, primscale,
                                               tab, sph);
    if (useT)
        transpose_tpl_kernel<<<(KPRIM * VOXK + 255) / 256, 256, 0, stream>>>(tpl, tplT);
    if (useMask)
        mask_kernel<<<NPIX / 16, 32, 0, stream>>>(raydir, sph, mask);

    raymarch_kernel<<<NPIX / 64, 64, 0, stream>>>(
        viewpos, raydir, dtp,
        useTdm ? tab : (const float*)nullptr,
        tplT, tpl,
        primpos, primrot, primscale,
        mask,
        (float*)d_out, useT, useTdm, useMask);
}